// AddictiveAttention_58686433132642
// MI455X (gfx1250) — compile-verified
//
#include <hip/hip_runtime.h>
#include <math.h>

// ---------------------------------------------------------------------------
// Additive attention, MI455X (gfx1250, wave32).
//   Stage 1: qp = Q@Wq, kp = K@Wk           (fp32 WMMA 16x16x4)
//   Stage 2: scores[b,q,k] = sum_h wv[h]*tanh(qp[b,q,h]+kp[b,k,h])  (v_tanh_f32)
//   Stage 3: masked softmax over k (fill -1e6)
//   Stage 4: out = attn @ V                  (fp32 WMMA 16x16x4)
// ---------------------------------------------------------------------------

typedef float v2f __attribute__((ext_vector_type(2)));
typedef float v8f __attribute__((ext_vector_type(8)));

constexpr int B  = 4;
constexpr int NQ = 256;
constexpr int NK = 1024;
constexpr int D  = 256;
constexpr int H  = 256;
constexpr float NEGF = -1e6f;

// __HIP_DEVICE_COMPILE__ is defined ONLY in the device pass. (The host pass
// predefines aux-target macros like __AMDGCN__ but __has_builtin() cannot see
// aux-target builtins there, so any host-side probe is meaningless.)
#if defined(__HIP_DEVICE_COMPILE__) && !__has_builtin(__builtin_amdgcn_wmma_f32_16x16x4_f32)
#error "gfx1250 device pass: __builtin_amdgcn_wmma_f32_16x16x4_f32 not available"
#endif

__device__ __forceinline__ float gpu_tanh(float x) {
#if defined(__HIP_DEVICE_COMPILE__)
#if __has_builtin(__builtin_amdgcn_tanhf)
  return __builtin_amdgcn_tanhf(x);
#else
  float r;
  asm("v_tanh_f32 %0, %1\n\tv_nop" : "=v"(r) : "v"(x));
  return r;
#endif
#else
  return tanhf(x);   // host parse only; never executed
#endif
}

__device__ __forceinline__ float gpu_exp2(float x) {
#if defined(__HIP_DEVICE_COMPILE__) && __has_builtin(__builtin_amdgcn_exp2f)
  return __builtin_amdgcn_exp2f(x);
#else
  return exp2f(x);
#endif
}

// ---------------------------------------------------------------------------
// Batched fp32 GEMM: Y[M,N] = A[M,K] @ Bm[K,N], one 16x16 tile per wave32.
// WMMA f32 16x16x4 layouts (ISA 7.12.2):
//   A 16x4 : lane L holds A[L&15][2*(L>>4) + j], j = vgpr index 0..1
//   B 4x16 : lane L holds B[2*(L>>4) + j][L&15]
//   C/D    : vgpr r, lane L -> C[r + 8*(L>>4)][L&15]
// ---------------------------------------------------------------------------
__global__ __launch_bounds__(256)
void gemm_wmma_f32(const float* __restrict__ A, const float* __restrict__ Bm,
                   float* __restrict__ Y, int M, int K, int N,
                   long long sA, long long sB, long long sY)
{
  const int lane = threadIdx.x & 31;
  const int wib  = threadIdx.x >> 5;             // wave in block (8 waves)
  const int tilesN = N >> 4;
  const int gw = blockIdx.x * 8 + wib;
  const int tileM = gw / tilesN;
  const int tileN = gw - tileM * tilesN;
  const int batch = blockIdx.y;

  A  += (long long)batch * sA;
  Bm += (long long)batch * sB;
  Y  += (long long)batch * sY;

  const int lo16  = lane & 15;
  const int kHalf = (lane >> 4) * 2;             // 0 or 2

  const float* aPtr = A + (long long)(tileM * 16 + lo16) * K + kHalf;
  const float* bPtr = Bm + (long long)kHalf * N + (tileN * 16 + lo16);

  v8f c = {};
#pragma unroll 4
  for (int kb = 0; kb < K; kb += 4) {
    v2f a = *(const v2f*)aPtr;                   // K = kb+kHalf, kb+kHalf+1
    v2f b;
    b.x = bPtr[0];
    b.y = bPtr[N];
    c = __builtin_amdgcn_wmma_f32_16x16x4_f32(
        /*neg_a=*/false, a, /*neg_b=*/false, b,
        /*c_mod=*/(short)0, c, /*reuse_a=*/false, /*reuse_b=*/false);
    aPtr += 4;
    bPtr += 4LL * N;
  }

  float* yPtr = Y + (long long)(tileM * 16 + (lane >> 4) * 8) * N
                  + (tileN * 16 + lo16);
#pragma unroll
  for (int r = 0; r < 8; ++r)
    yPtr[(long long)r * N] = c[r];
}

// ---------------------------------------------------------------------------
// scores[b, q0+qi, k0+ki] = sum_h wv[h] * tanh(qp[b,q0+qi,h] + kp[b,k0+ki,h])
// 16x16 (q,k) tile per block of 256 threads; tiles staged in LDS with
// stride 260 (mod 64 == 4) so the 16 distinct k-rows hit 16 distinct banks.
// ---------------------------------------------------------------------------
__global__ __launch_bounds__(256)
void scores_tanh_kernel(const float* __restrict__ qp, const float* __restrict__ kp,
                        const float* __restrict__ wv, float* __restrict__ scores)
{
  constexpr int HS = H + 4;                      // 260: bank-conflict-free
  __shared__ float qs[16 * HS];
  __shared__ float ks[16 * HS];
  __shared__ float wvs[H];

  const int t  = threadIdx.x;
  const int b  = blockIdx.z;
  const int q0 = blockIdx.y * 16;
  const int k0 = blockIdx.x * 16;

  // Stage tiles: 16 rows x 256 floats = 1024 float4 each; 4 float4 per thread.
  const float4* qg = (const float4*)(qp + ((long long)b * NQ + q0) * H);
  const float4* kg = (const float4*)(kp + ((long long)b * NK + k0) * H);
#pragma unroll
  for (int i = 0; i < 4; ++i) {
    const int f    = t + i * 256;                // 0..1023
    const int row  = f >> 6;                     // /64 float4 per row
    const int col4 = f & 63;
    float4 qv = qg[row * 64 + col4];
    float4 kv = kg[row * 64 + col4];
    *(float4*)&qs[row * HS + col4 * 4] = qv;
    *(float4*)&ks[row * HS + col4 * 4] = kv;
  }
  wvs[t] = wv[t];                                // H == blockDim.x == 256
  __syncthreads();

  const int qi = t >> 4;
  const int ki = t & 15;
  const float* qrow = &qs[qi * HS];
  const float* krow = &ks[ki * HS];

  float acc = 0.f;
#pragma unroll 8
  for (int h = 0; h < H; ++h) {
    acc = fmaf(wvs[h], gpu_tanh(qrow[h] + krow[h]), acc);
  }
  scores[((long long)b * NQ + q0 + qi) * NK + (k0 + ki)] = acc;
}

// ---------------------------------------------------------------------------
// In-place masked softmax over the last axis (NK=1024), one row per block.
// Matches reference: scores = where(k < valid_len, s, -1e6); softmax.
// ---------------------------------------------------------------------------
__global__ __launch_bounds__(256)
void softmax_mask_kernel(float* __restrict__ attn, const int* __restrict__ valid_lens)
{
  const int row = blockIdx.x;                    // 0 .. B*NQ-1
  const int b   = row / NQ;
  const int vl  = valid_lens[b];
  float* srow   = attn + (long long)row * NK;
  const int t   = threadIdx.x;

  __shared__ float redMax[8];
  __shared__ float redSum[8];

  float vals[4];
  float m = -__builtin_inff();
#pragma unroll
  for (int i = 0; i < 4; ++i) {
    const int k = t + i * 256;
    float s = srow[k];
    s = (k < vl) ? s : NEGF;
    vals[i] = s;
    m = fmaxf(m, s);
  }
  for (int off = 16; off > 0; off >>= 1)
    m = fmaxf(m, __shfl_xor(m, off, 32));
  if ((t & 31) == 0) redMax[t >> 5] = m;
  __syncthreads();
  float mAll = redMax[0];
#pragma unroll
  for (int i = 1; i < 8; ++i) mAll = fmaxf(mAll, redMax[i]);

  constexpr float LOG2E = 1.4426950408889634f;
  float ssum = 0.f;
#pragma unroll
  for (int i = 0; i < 4; ++i) {
    vals[i] = gpu_exp2((vals[i] - mAll) * LOG2E);
    ssum += vals[i];
  }
  for (int off = 16; off > 0; off >>= 1)
    ssum += __shfl_xor(ssum, off, 32);
  if ((t & 31) == 0) redSum[t >> 5] = ssum;
  __syncthreads();
  float total = 0.f;
#pragma unroll
  for (int i = 0; i < 8; ++i) total += redSum[i];

  const float inv = 1.f / total;
#pragma unroll
  for (int i = 0; i < 4; ++i)
    srow[t + i * 256] = vals[i] * inv;
}

// ---------------------------------------------------------------------------
extern "C" void kernel_launch(void* const* d_in, const int* in_sizes, int n_in,
                              void* d_out, int out_size, void* d_ws, size_t ws_size,
                              hipStream_t stream) {
  const float* queries = (const float*)d_in[0];  // [B,NQ,D]
  const float* keys    = (const float*)d_in[1];  // [B,NK,D]
  const float* values  = (const float*)d_in[2];  // [B,NK,D]
  const int*   vlens   = (const int*)  d_in[3];  // [B]
  const float* Wq      = (const float*)d_in[4];  // [D,H]
  const float* Wk      = (const float*)d_in[5];  // [D,H]
  const float* w_v     = (const float*)d_in[6];  // [H]
  float* out = (float*)d_out;                    // [B,NQ,D]

  float* qp   = (float*)d_ws;                    // B*NQ*H  (1 MB)
  float* kp   = qp + (size_t)B * NQ * H;         // B*NK*H  (4 MB)
  float* attn = kp + (size_t)B * NK * H;         // B*NQ*NK (4 MB)

  const dim3 blk(256);

  // Stage 1: projections (batch dim folds into M since matmul is over last axis).
  gemm_wmma_f32<<<dim3(((B * NQ / 16) * (H / 16)) / 8, 1), blk, 0, stream>>>(
      queries, Wq, qp, B * NQ, D, H, 0, 0, 0);
  gemm_wmma_f32<<<dim3(((B * NK / 16) * (H / 16)) / 8, 1), blk, 0, stream>>>(
      keys, Wk, kp, B * NK, D, H, 0, 0, 0);

  // Stage 2: additive scores with hardware tanh (the roofline bottleneck).
  scores_tanh_kernel<<<dim3(NK / 16, NQ / 16, B), blk, 0, stream>>>(
      qp, kp, w_v, attn);

  // Stage 3: masked softmax (in place).
  softmax_mask_kernel<<<dim3(B * NQ), blk, 0, stream>>>(attn, vlens);

  // Stage 4: out = attn @ V (batched).
  gemm_wmma_f32<<<dim3(((NQ / 16) * (D / 16)) / 8, B), blk, 0, stream>>>(
      attn, values, out, NQ, NK, D,
      (long long)NQ * NK, (long long)NK * D, (long long)NQ * D);
}